// HespiasLoss_35871566856392
// MI455X (gfx1250) — compile-verified
//
#include <hip/hip_runtime.h>
#include <hip/hip_bf16.h>

#define ORDER_COUNT   64
#define N_LOGITS      16960
#define BATCH         4096
#define EPS           0.1f
#define WAVES_PER_BLK 8
#define NBLOCKS       (BATCH / WAVES_PER_BLK)   // 512 partials

typedef __attribute__((ext_vector_type(2))) float v2f;
typedef __attribute__((ext_vector_type(8))) float v8f;

// One-instruction wave32 butterfly step: DS_SWIZZLE_B32 group-of-32 xor mode.
// offset = (xor_mask << 10) | 0x1F  (and_mask=0x1F, or_mask=0)
template <int IMM>
__device__ __forceinline__ float swz_xor(float v) {
    return __int_as_float(__builtin_amdgcn_ds_swizzle(__float_as_int(v), IMM));
}

__device__ __forceinline__ float wave_sum32(float v) {
    v += swz_xor<0x401F>(v);   // SWAPX16
    v += swz_xor<0x201F>(v);   // SWAPX8
    v += swz_xor<0x101F>(v);   // SWAPX4
    v += swz_xor<0x081F>(v);   // SWAPX2
    v += swz_xor<0x041F>(v);   // SWAPX1
    return v;
}
__device__ __forceinline__ float wave_max32(float v) {
    v = fmaxf(v, swz_xor<0x401F>(v));
    v = fmaxf(v, swz_xor<0x201F>(v));
    v = fmaxf(v, swz_xor<0x101F>(v));
    v = fmaxf(v, swz_xor<0x081F>(v));
    v = fmaxf(v, swz_xor<0x041F>(v));
    return v;
}

// One wave32 per batch sample; 8 waves (256 threads) per block.
__global__ __launch_bounds__(256) void loss_partials(
    const float* __restrict__ pred,    // [B, N_LOGITS]
    const int*   __restrict__ tgt,     // [B]
    const int*   __restrict__ c2o,     // [N_CAT]
    const int*   __restrict__ c2f,     // [N_CAT]
    const int*   __restrict__ fstart,  // [64]
    const int*   __restrict__ fend,    // [64]
    const int*   __restrict__ findex,  // [512]
    const int*   __restrict__ cstart,  // [512]
    const int*   __restrict__ cend,    // [512]
    const int*   __restrict__ cindex,  // [N_CAT]
    float*       __restrict__ partials)
{
    __shared__ float smem[WAVES_PER_BLK];
    const int lane = threadIdx.x & 31;
    const int wv   = threadIdx.x >> 5;
    const int b    = blockIdx.x * WAVES_PER_BLK + wv;
    const float* row = pred + (size_t)b * N_LOGITS;

    const int tc  = tgt[b];
    const int ord = c2o[tc];
    const int fam = c2f[tc];

    // ---------- order CE over row[0:64]: one b64 load per lane ----------
    const v2f x01 = *(const v2f*)(row + 2 * lane);   // 8B-aligned (row stride mult. of 8B)
    const float x0 = x01.x, x1 = x01.y;
    float m  = wave_max32(fmaxf(x0, x1));
    float ex = __expf(x0 - m) + __expf(x1 - m);
    float sx = x0 + x1;
    ex = wave_sum32(ex);
    sx = wave_sum32(sx);
    float lse_o    = m + __logf(ex);
    float lt_o     = row[ord];                                   // uniform per wave
    float order_ce = lse_o - ((1.0f - EPS) * lt_o + EPS * (sx * (1.0f / 64.0f)));

    // ---------- family CE over 8-wide segment ----------
    const int fs = fstart[ord];
    const int fn = fend[ord] - fs;                               // == 8
    float fx  = (lane < fn) ? row[fs + lane] : -__builtin_inff();
    float fm  = wave_max32(fx);
    float fex = (lane < fn) ? __expf(fx - fm) : 0.0f;
    float fsx = (lane < fn) ? fx : 0.0f;
    fex = wave_sum32(fex);
    fsx = wave_sum32(fsx);
    float lse_f  = fm + __logf(fex);
    float lt_f   = row[fs + findex[fam]];
    float fam_ce = lse_f - ((1.0f - EPS) * lt_f + EPS * (fsx / (float)fn));

    // ---------- category CE over 32-wide segment ----------
    const int cs = cstart[fam];
    const int cn = cend[fam] - cs;                               // == 32
    float cx  = (lane < cn) ? row[cs + lane] : -__builtin_inff();
    float cm  = wave_max32(cx);
    float cex = (lane < cn) ? __expf(cx - cm) : 0.0f;
    float csx = (lane < cn) ? cx : 0.0f;
    cex = wave_sum32(cex);
    csx = wave_sum32(csx);
    float lse_c  = cm + __logf(cex);
    float lt_c   = row[cs + cindex[tc]];
    float cat_ce = lse_c - ((1.0f - EPS) * lt_c + EPS * (csx / (float)cn));

    const float total = order_ce + fam_ce + cat_ce;
    if (lane == 0) smem[wv] = total;
    __syncthreads();
    if (threadIdx.x == 0) {
        float acc = 0.0f;
#pragma unroll
        for (int i = 0; i < WAVES_PER_BLK; ++i) acc += smem[i];
        partials[blockIdx.x] = acc;                              // deterministic order
    }
}

// Single-wave finisher: sum 512 partials exactly in f32 with
// V_WMMA_F32_16X16X4_F32, A = all-ones. D[i][j] = sum_k B[k][j] independent
// of i and of B's lane->(k,j) mapping, so B can be loaded contiguously
// (one b64 per lane per chunk). Summing the first C/D VGPR over all 32
// lanes (rows M=0 and M=8, both equal to the column sums) gives 2x the total.
__global__ __launch_bounds__(32) void final_reduce(
    const float* __restrict__ partials, float* __restrict__ out)
{
    const int lane = threadIdx.x;  // exactly one wave32, EXEC all ones
    v2f a;
    a.x = 1.0f;
    a.y = 1.0f;
    v8f c = {};
#pragma unroll
    for (int chunk = 0; chunk < 8; ++chunk) {
        const v2f bv = *(const v2f*)(partials + chunk * 64 + 2 * lane);
        c = __builtin_amdgcn_wmma_f32_16x16x4_f32(
                /*neg_a=*/false, a, /*neg_b=*/false, bv,
                /*c_mod=*/(short)0, c, /*reuse_a=*/false, /*reuse_b=*/false);
    }
    float t = wave_sum32(c[0]) * 0.5f;        // undo the M=0 / M=8 duplication
    if (lane == 0) out[0] = t * (1.0f / (float)BATCH);
}

extern "C" void kernel_launch(void* const* d_in, const int* in_sizes, int n_in,
                              void* d_out, int out_size, void* d_ws, size_t ws_size,
                              hipStream_t stream) {
    const float* pred   = (const float*)d_in[0];
    const int*   tgt    = (const int*)d_in[1];
    const int*   c2o    = (const int*)d_in[2];
    const int*   c2f    = (const int*)d_in[3];
    const int*   fstart = (const int*)d_in[4];
    const int*   fend   = (const int*)d_in[5];
    const int*   findex = (const int*)d_in[6];
    const int*   cstart = (const int*)d_in[7];
    const int*   cend   = (const int*)d_in[8];
    const int*   cindex = (const int*)d_in[9];
    float* partials = (float*)d_ws;            // 512 floats, fully overwritten

    loss_partials<<<NBLOCKS, 256, 0, stream>>>(pred, tgt, c2o, c2f, fstart, fend,
                                               findex, cstart, cend, cindex,
                                               partials);
    final_reduce<<<1, 32, 0, stream>>>(partials, (float*)d_out);
}